// Mymodel_35691178230213
// MI455X (gfx1250) — compile-verified
//
#include <hip/hip_runtime.h>
#include <math.h>

typedef float v2f __attribute__((ext_vector_type(2)));
typedef float v8f __attribute__((ext_vector_type(8)));

#define NPTS 2048
#define BATCH 8
#define KNN 20
#define NEG_INF (-3.402823e38f)

#define WMMA_F32(A_, B_, C_) \
    __builtin_amdgcn_wmma_f32_16x16x4_f32(false, (A_), false, (B_), (short)0, (C_), false, false)

// ---------------------------------------------------------------------------
// Specialized TN GEMM, fp32 WMMA:  Out[b][m][n] = sum_k A[m,k] * B[b][k,n]
//   A: row-major weights [M, Kd]          (k-stride 1, m-stride Kd)
//   B: activations, k-stride NPTS, n-stride 1, batch stride bsB
// One wave computes a 32x32 tile (2 A frags x 2 B frags -> 4 accumulators).
// A-row addresses are CLAMPED (not predicated): garbage rows only feed D rows
// that are never stored.  k-loop: unguarded main + guarded tail.
// EVENK: A fragment loaded as one b64 (alignment provable when Kd is even).
// ---------------------------------------------------------------------------
template <bool EVENK>
__global__ void gemm_wmma(const float* __restrict__ A, const float* __restrict__ B,
                          float* __restrict__ Out,
                          int M, int N, int Kd, long bsB, long bsO)
{
    int mt = (M + 31) >> 5, nt = N >> 5;
    int tile = blockIdx.x * 4 + (threadIdx.x >> 5);
    if (tile >= mt * nt) return;
    int tm = tile / nt, tn = tile % nt;
    int m0 = tm * 32, n0 = tn * 32;
    const float* Bb = B + (long)blockIdx.y * bsB;
    float* Ob = Out + (long)blockIdx.y * bsO;

    int lane = threadIdx.x & 31;
    int half = lane >> 4;
    int l    = lane & 15;

    int mA0 = m0 + l;       if (mA0 > M - 1) mA0 = M - 1;
    int mA1 = m0 + 16 + l;  if (mA1 > M - 1) mA1 = M - 1;

    const float* pa0 = A  + (long)mA0 * Kd + 2 * half;
    const float* pa1 = A  + (long)mA1 * Kd + 2 * half;
    const float* pb0 = Bb + (long)(2 * half) * NPTS + (n0 + l);
    const float* pb1 = pb0 + 16;

    v8f acc00 = {0.f,0.f,0.f,0.f,0.f,0.f,0.f,0.f};
    v8f acc01 = acc00, acc10 = acc00, acc11 = acc00;

    int kmain = Kd & ~3;
    for (int k0 = 0; k0 < kmain; k0 += 4) {
        v2f a0, a1, b0, b1;
        if (EVENK) {
            a0 = *(const v2f*)pa0;
            a1 = *(const v2f*)pa1;
        } else {
            a0.x = pa0[0]; a0.y = pa0[1];
            a1.x = pa1[0]; a1.y = pa1[1];
        }
        b0.x = pb0[0]; b0.y = pb0[NPTS];
        b1.x = pb1[0]; b1.y = pb1[NPTS];
        acc00 = WMMA_F32(a0, b0, acc00);
        acc01 = WMMA_F32(a0, b1, acc01);
        acc10 = WMMA_F32(a1, b0, acc10);
        acc11 = WMMA_F32(a1, b1, acc11);
        pa0 += 4; pa1 += 4;
        pb0 += 4L * NPTS; pb1 += 4L * NPTS;
    }
    if (kmain < Kd) {                       // zero-padded k tail (K=3 / K=6 / K%4==2)
        int ka = kmain + 2 * half;
        bool g0 = ka < Kd, g1 = ka + 1 < Kd;
        v2f a0, a1, b0, b1;
        a0.x = g0 ? pa0[0]    : 0.f;  a0.y = g1 ? pa0[1]    : 0.f;
        a1.x = g0 ? pa1[0]    : 0.f;  a1.y = g1 ? pa1[1]    : 0.f;
        b0.x = g0 ? pb0[0]    : 0.f;  b0.y = g1 ? pb0[NPTS] : 0.f;
        b1.x = g0 ? pb1[0]    : 0.f;  b1.y = g1 ? pb1[NPTS] : 0.f;
        acc00 = WMMA_F32(a0, b0, acc00);
        acc01 = WMMA_F32(a0, b1, acc01);
        acc10 = WMMA_F32(a1, b0, acc10);
        acc11 = WMMA_F32(a1, b1, acc11);
    }
#pragma unroll
    for (int r = 0; r < 8; ++r) {
        int row0 = m0 + r + 8 * half;
        if (row0 < M) {
            Ob[(long)row0 * N + n0 + l]      = acc00[r];
            Ob[(long)row0 * N + n0 + 16 + l] = acc01[r];
        }
        int row1 = m0 + 16 + r + 8 * half;
        if (row1 < M) {
            Ob[(long)row1 * N + n0 + l]      = acc10[r];
            Ob[(long)row1 * N + n0 + 16 + l] = acc11[r];
        }
    }
}

// ---------------------------------------------------------------------------
// Per-point squared norms: sq[b][n] = sum_c x[b,c,n]^2
// ---------------------------------------------------------------------------
__global__ void row_sqnorm(const float* __restrict__ X, long bsX,
                           float* __restrict__ sq, int C)
{
    int n = blockIdx.x * blockDim.x + threadIdx.x;
    int bb = blockIdx.y;
    if (n >= NPTS) return;
    const float* Xb = X + (long)bb * bsX;
    float s = 0.f;
    for (int c = 0; c < C; ++c) { float v = Xb[(long)c * NPTS + n]; s += v * v; }
    sq[(long)bb * NPTS + n] = s;
}

// ---------------------------------------------------------------------------
// Fused Gram + top-K:  pd[n,m] = 2*x_n.x_m - |x_n|^2 - |x_m|^2, kNN per row.
// One wave per 16-row strip; each iteration computes a 16x32 distance slab
// (shared A fragment, 2 WMMA) -> LDS -> per-row insertion lists in VGPRs.
// pd never touches HBM (saves ~537 MB of round-trip at 23.3 TB/s).
// ---------------------------------------------------------------------------
__global__ void knn_gram_topk(const float* __restrict__ X, long bsX,
                              const float* __restrict__ sq,
                              int* __restrict__ idx, int C)
{
    __shared__ float tile[16 * 33];
    int bb = blockIdx.y;
    int r0 = blockIdx.x * 16;
    const float* Xb  = X  + (long)bb * bsX;
    const float* sqb = sq + (long)bb * NPTS;

    int lane = threadIdx.x & 31;
    int half = lane >> 4;
    int l    = lane & 15;

    float sqrow[8];
#pragma unroll
    for (int r = 0; r < 8; ++r) sqrow[r] = sqb[r0 + r + 8 * half];

    float tv[KNN]; int ti[KNN];
#pragma unroll
    for (int j = 0; j < KNN; ++j) { tv[j] = NEG_INF; ti[j] = 0; }
    int grow = r0 + lane;            // selection row for lanes 0..15

    const float* pa = Xb + (long)(2 * half) * NPTS + (r0 + l);
    int kmain = C & ~3;
    const long step = 4L * NPTS;

    for (int ct = 0; ct < NPTS / 32; ++ct) {
        int n0 = ct * 32;
        const float* qa  = pa;
        const float* qb0 = Xb + (long)(2 * half) * NPTS + (n0 + l);
        const float* qb1 = qb0 + 16;

        v8f acc0 = {0.f,0.f,0.f,0.f,0.f,0.f,0.f,0.f};
        v8f acc1 = acc0;
        for (int k0 = 0; k0 < kmain; k0 += 4) {
            v2f a, b0, b1;
            a.x  = qa[0];   a.y  = qa[NPTS];
            b0.x = qb0[0];  b0.y = qb0[NPTS];
            b1.x = qb1[0];  b1.y = qb1[NPTS];
            acc0 = WMMA_F32(a, b0, acc0);
            acc1 = WMMA_F32(a, b1, acc1);
            qa += step; qb0 += step; qb1 += step;
        }
        if (kmain < C) {             // C == 3 tail
            int ka = kmain + 2 * half;
            bool g0 = ka < C, g1 = ka + 1 < C;
            v2f a, b0, b1;
            a.x  = g0 ? qa[0]    : 0.f;  a.y  = g1 ? qa[NPTS]  : 0.f;
            b0.x = g0 ? qb0[0]   : 0.f;  b0.y = g1 ? qb0[NPTS] : 0.f;
            b1.x = g0 ? qb1[0]   : 0.f;  b1.y = g1 ? qb1[NPTS] : 0.f;
            acc0 = WMMA_F32(a, b0, acc0);
            acc1 = WMMA_F32(a, b1, acc1);
        }

        float sqc0 = sqb[n0 + l];
        float sqc1 = sqb[n0 + 16 + l];
#pragma unroll
        for (int r = 0; r < 8; ++r) {
            int row = r + 8 * half;
            tile[row * 33 + l]      = 2.f * acc0[r] - sqrow[r] - sqc0;
            tile[row * 33 + 16 + l] = 2.f * acc1[r] - sqrow[r] - sqc1;
        }
        __syncthreads();

        if (lane < 16) {
            for (int j = 0; j < 32; ++j) {
                int col = n0 + j;
                if (col == grow) continue;               // exclude self
                float v = tile[lane * 33 + j];
                if (v > tv[KNN - 1]) {
                    int ci = col;
#pragma unroll
                    for (int t = 0; t < KNN; ++t) {
                        if (v > tv[t]) {
                            float fv = tv[t]; int fi = ti[t];
                            tv[t] = v; ti[t] = ci; v = fv; ci = fi;
                        }
                    }
                }
            }
        }
        __syncthreads();
    }
    if (lane < 16) {
#pragma unroll
        for (int j = 0; j < KNN; ++j)
            idx[(long)bb * NPTS * KNN + (long)grow * KNN + j] = ti[j];
    }
}

// ---------------------------------------------------------------------------
// Attention gather phase. One wave32 per point; lanes span the E dimension.
// scores_j = softmax( q . XK[:,idx_j] / sqrt(E) );  out = sum_j s_j (XV_j - XV_n)
// ---------------------------------------------------------------------------
__global__ void attention_kernel(const float* __restrict__ XQ,
                                 const float* __restrict__ XK,
                                 const float* __restrict__ XV,
                                 const int* __restrict__ idx,
                                 float* __restrict__ outp, int E)
{
    int wave = (blockIdx.x * blockDim.x + threadIdx.x) >> 5;
    int lane = threadIdx.x & 31;
    int bb = wave / NPTS, n = wave % NPTS;
    long base = (long)bb * E * NPTS;
    const float* Q  = XQ + base;
    const float* K_ = XK + base;
    const float* V_ = XV + base;
    float* O = outp + base;
    const int* id = idx + (long)bb * NPTS * KNN + (long)n * KNN;
    int NE = (E + 31) >> 5;
    float rscale = rsqrtf((float)E);

    float qv[4];
#pragma unroll
    for (int i = 0; i < 4; ++i) {
        int e = lane + 32 * i;
        qv[i] = (i < NE && e < E) ? Q[(long)e * NPTS + n] : 0.f;
    }

    int nj[KNN];
#pragma unroll
    for (int j = 0; j < KNN; ++j) nj[j] = id[j];

    float sc[KNN];
#pragma unroll
    for (int j = 0; j < KNN; ++j) {
        float p = 0.f;
#pragma unroll
        for (int i = 0; i < 4; ++i) {
            int e = lane + 32 * i;
            float kv = (i < NE && e < E) ? K_[(long)e * NPTS + nj[j]] : 0.f;
            p += qv[i] * kv;
        }
#pragma unroll
        for (int m = 16; m >= 1; m >>= 1) p += __shfl_xor(p, m, 32);
        sc[j] = p * rscale;
    }
    float mx = sc[0];
#pragma unroll
    for (int j = 1; j < KNN; ++j) mx = fmaxf(mx, sc[j]);
    float s = 0.f;
#pragma unroll
    for (int j = 0; j < KNN; ++j) { sc[j] = __expf(sc[j] - mx); s += sc[j]; }
    float inv = 1.f / s;
#pragma unroll
    for (int j = 0; j < KNN; ++j) sc[j] *= inv;

#pragma unroll
    for (int i = 0; i < 4; ++i) {
        int e = lane + 32 * i;
        if (i < NE && e < E) {
            float xvn = V_[(long)e * NPTS + n];
            float o = 0.f;
#pragma unroll
            for (int j = 0; j < KNN; ++j)
                o += sc[j] * (V_[(long)e * NPTS + nj[j]] - xvn);
            O[(long)e * NPTS + n] = o;
        }
    }
}

// ---------------------------------------------------------------------------
// BatchNorm stats over (B, N) per channel (training-mode, biased var).
// ---------------------------------------------------------------------------
__global__ void bn_stats(const float* __restrict__ pre, int C,
                         float* __restrict__ meanb, float* __restrict__ rstdb)
{
    int c = blockIdx.x;
    __shared__ float s1[256], s2[256];
    float a = 0.f, b = 0.f;
    for (int bb = 0; bb < BATCH; ++bb) {
        const float* row = pre + ((long)bb * C + c) * NPTS;
        for (int n = threadIdx.x; n < NPTS; n += 256) {
            float v = row[n];
            a += v; b += v * v;
        }
    }
    s1[threadIdx.x] = a; s2[threadIdx.x] = b;
    __syncthreads();
    for (int s = 128; s > 0; s >>= 1) {
        if (threadIdx.x < s) { s1[threadIdx.x] += s1[threadIdx.x + s];
                               s2[threadIdx.x] += s2[threadIdx.x + s]; }
        __syncthreads();
    }
    if (threadIdx.x == 0) {
        float cnt = (float)(BATCH * NPTS);
        float m = s1[0] / cnt;
        float var = s2[0] / cnt - m * m;
        meanb[c] = m;
        rstdb[c] = rsqrtf(var + 1e-5f);
    }
}

// Normalize + LeakyReLU, scatter into dst[b*dstB + (coff+c)*N + n]
__global__ void bn_apply(const float* __restrict__ pre,
                         const float* __restrict__ meanb, const float* __restrict__ rstdb,
                         float* __restrict__ dst, int C, long dstB, long coff)
{
    long i = (long)blockIdx.x * blockDim.x + threadIdx.x;
    long tot = (long)BATCH * C * NPTS;
    if (i >= tot) return;
    int n = (int)(i % NPTS);
    long t = i / NPTS;
    int c = (int)(t % C);
    int bb = (int)(t / C);
    float v = (pre[i] - meanb[c]) * rstdb[c];
    v = v >= 0.f ? v : 0.2f * v;
    dst[(long)bb * dstB + (coff + c) * (long)NPTS + n] = v;
}

// Copy [B,C,N] slice into dst with channel offset (the concat([res, x]) tail)
__global__ void copy_cn(const float* __restrict__ src, long srcB,
                        float* __restrict__ dst, long dstB, long coff, int C)
{
    long i = (long)blockIdx.x * blockDim.x + threadIdx.x;
    long tot = (long)BATCH * C * NPTS;
    if (i >= tot) return;
    int n = (int)(i % NPTS);
    long t = i / NPTS;
    int c = (int)(t % C);
    int bb = (int)(t / C);
    dst[(long)bb * dstB + (coff + c) * (long)NPTS + n] =
        src[(long)bb * srcB + (long)c * NPTS + n];
}

// Fused BN + LeakyReLU + max/mean pooling over N for x5 (C=1024).
__global__ void pool_maxmean(const float* __restrict__ pre,
                             const float* __restrict__ meanb, const float* __restrict__ rstdb,
                             float* __restrict__ p)
{
    int c = blockIdx.x, bb = blockIdx.y;
    __shared__ float sm[256], ss[256];
    float mx = NEG_INF, sum = 0.f;
    float m = meanb[c], rs = rstdb[c];
    const float* row = pre + ((long)bb * 1024 + c) * NPTS;
    for (int n = threadIdx.x; n < NPTS; n += 256) {
        float v = (row[n] - m) * rs;
        v = v >= 0.f ? v : 0.2f * v;
        mx = fmaxf(mx, v); sum += v;
    }
    sm[threadIdx.x] = mx; ss[threadIdx.x] = sum;
    __syncthreads();
    for (int s = 128; s > 0; s >>= 1) {
        if (threadIdx.x < s) { sm[threadIdx.x] = fmaxf(sm[threadIdx.x], sm[threadIdx.x + s]);
                               ss[threadIdx.x] += ss[threadIdx.x + s]; }
        __syncthreads();
    }
    if (threadIdx.x == 0) {
        p[(long)bb * 2048 + c]        = sm[0];
        p[(long)bb * 2048 + 1024 + c] = ss[0] / (float)NPTS;
    }
}

// Linear + bias + BN over batch(8) + LeakyReLU, one feature per block.
__global__ void lin_bn(const float* __restrict__ in, const float* __restrict__ W,
                       const float* __restrict__ bias, float* __restrict__ out,
                       int Kd, int F)
{
    int f = blockIdx.x;
    __shared__ float red[256];
    float y[8];
#pragma unroll
    for (int bb = 0; bb < 8; ++bb) {
        float a = 0.f;
        for (int k = threadIdx.x; k < Kd; k += 256)
            a += in[(long)bb * Kd + k] * W[(long)f * Kd + k];
        red[threadIdx.x] = a;
        __syncthreads();
        for (int s = 128; s > 0; s >>= 1) {
            if (threadIdx.x < s) red[threadIdx.x] += red[threadIdx.x + s];
            __syncthreads();
        }
        y[bb] = red[0] + bias[f];
        __syncthreads();
    }
    if (threadIdx.x == 0) {
        float m = 0.f;
#pragma unroll
        for (int bb = 0; bb < 8; ++bb) m += y[bb];
        m *= 0.125f;
        float var = 0.f;
#pragma unroll
        for (int bb = 0; bb < 8; ++bb) { float d = y[bb] - m; var += d * d; }
        var *= 0.125f;
        float rs = rsqrtf(var + 1e-5f);
#pragma unroll
        for (int bb = 0; bb < 8; ++bb) {
            float z = (y[bb] - m) * rs;
            z = z >= 0.f ? z : 0.2f * z;
            out[(long)bb * F + f] = z;
        }
    }
}

__global__ void lin_final(const float* __restrict__ in, const float* __restrict__ W,
                          const float* __restrict__ bias, float* __restrict__ out)
{
    int t = blockIdx.x * blockDim.x + threadIdx.x;
    if (t >= 8 * 40) return;
    int bb = t / 40, f = t % 40;
    float a = bias[f];
    for (int k = 0; k < 256; ++k)
        a += in[(long)bb * 256 + k] * W[(long)f * 256 + k];
    out[t] = a;
}

// ---------------------------------------------------------------------------
// Host orchestration
// ---------------------------------------------------------------------------
extern "C" void kernel_launch(void* const* d_in, const int* in_sizes, int n_in,
                              void* d_out, int out_size, void* d_ws, size_t ws_size,
                              hipStream_t stream)
{
    (void)in_sizes; (void)n_in; (void)out_size; (void)ws_size;

    const float* x0 = (const float*)d_in[0];
    const float* saw[4][4];
    for (int s = 0; s < 4; ++s)
        for (int w = 0; w < 4; ++w)
            saw[s][w] = (const float*)d_in[1 + s * 4 + w];
    const float* convw[5];
    for (int i = 0; i < 5; ++i) convw[i] = (const float*)d_in[17 + i];
    const float* lin1_w = (const float*)d_in[22];
    const float* lin1_b = (const float*)d_in[23];
    const float* lin2_w = (const float*)d_in[24];
    const float* lin2_b = (const float*)d_in[25];
    const float* lin3_w = (const float*)d_in[26];
    const float* lin3_b = (const float*)d_in[27];

    float* ws = (float*)d_ws;
    size_t off = 0;
    auto arena = [&](size_t n) { float* q = ws + off; off += (n + 3) & ~(size_t)3; return q; };
    float* sq    = arena((size_t)BATCH * NPTS);
    float* XQ    = arena((size_t)BATCH * 128 * NPTS);
    float* XK    = arena((size_t)BATCH * 128 * NPTS);
    float* XV    = arena((size_t)BATCH * 128 * NPTS);
    float* att   = arena((size_t)BATCH * 128 * NPTS);
    float* pre   = arena((size_t)BATCH * 1024 * NPTS);
    float* hbuf  = arena((size_t)BATCH * 256 * NPTS);
    float* xc    = arena((size_t)BATCH * 512 * NPTS);
    float* meanb = arena(1024);
    float* rstdb = arena(1024);
    float* pbuf  = arena((size_t)BATCH * 2048);
    float* h1    = arena((size_t)BATCH * 512);
    float* h2    = arena((size_t)BATCH * 256);
    int*   idxb  = (int*)arena((size_t)BATCH * NPTS * KNN);

    auto gemm = [&](const float* A, const float* Bp, long bsB,
                    float* Outp, int M, int Nn, int Kd) {
        int mt = (M + 31) / 32, nt = Nn / 32;
        int tiles = mt * nt;
        dim3 g((tiles + 3) / 4, BATCH);
        if ((Kd & 1) == 0)
            gemm_wmma<true><<<g, 128, 0, stream>>>(A, Bp, Outp, M, Nn, Kd,
                                                   bsB, (long)M * Nn);
        else
            gemm_wmma<false><<<g, 128, 0, stream>>>(A, Bp, Outp, M, Nn, Kd,
                                                    bsB, (long)M * Nn);
    };

    auto conv_bn_to = [&](const float* hsrc, long bsH, int Cin, int O, const float* W,
                          float* dst, long dstB, long coff) {
        gemm(W, hsrc, bsH, pre, O, NPTS, Cin);
        bn_stats<<<O, 256, 0, stream>>>(pre, O, meanb, rstdb);
        long tot = (long)BATCH * O * NPTS;
        bn_apply<<<(int)((tot + 255) / 256), 256, 0, stream>>>(pre, meanb, rstdb,
                                                               dst, O, dstB, coff);
    };

    auto sa_block = [&](const float* xp, long bsX, int C, const float* const* w) {
        dim3 gs((NPTS + 255) / 256, BATCH);
        row_sqnorm<<<gs, 256, 0, stream>>>(xp, bsX, sq, C);
        knn_gram_topk<<<dim3(NPTS / 16, BATCH), 32, 0, stream>>>(xp, bsX, sq, idxb, C);
        gemm(w[0], xp, bsX, XQ, C, NPTS, C);
        gemm(w[1], xp, bsX, XK, C, NPTS, C);
        gemm(w[2], xp, bsX, XV, C, NPTS, C);
        attention_kernel<<<BATCH * NPTS / 8, 256, 0, stream>>>(XQ, XK, XV, idxb, att, C);
        gemm(w[3], att, (long)C * NPTS, pre, C, NPTS, C);
        bn_stats<<<C, 256, 0, stream>>>(pre, C, meanb, rstdb);
        long tot = (long)BATCH * C * NPTS;
        int nb = (int)((tot + 255) / 256);
        bn_apply<<<nb, 256, 0, stream>>>(pre, meanb, rstdb, hbuf, C, (long)2 * C * NPTS, 0);
        copy_cn<<<nb, 256, 0, stream>>>(xp, bsX, hbuf, (long)2 * C * NPTS, C, C);
    };

    // SA1 (C=3) -> conv1 -> x1 (64ch at xc+0)
    sa_block(x0, 3L * NPTS, 3, saw[0]);
    conv_bn_to(hbuf, 6L * NPTS, 6, 64, convw[0], xc, 512L * NPTS, 0);
    // SA2 (C=64) -> conv2 -> x2 (64ch at xc+64)
    sa_block(xc, 512L * NPTS, 64, saw[1]);
    conv_bn_to(hbuf, 128L * NPTS, 128, 64, convw[1], xc, 512L * NPTS, 64);
    // SA3 (C=64) -> conv3 -> x3 (128ch at xc+128)
    sa_block(xc + 64L * NPTS, 512L * NPTS, 64, saw[2]);
    conv_bn_to(hbuf, 128L * NPTS, 128, 128, convw[2], xc, 512L * NPTS, 128);
    // SA4 (C=128) -> conv4 -> x4 (256ch at xc+256)
    sa_block(xc + 128L * NPTS, 512L * NPTS, 128, saw[3]);
    conv_bn_to(hbuf, 256L * NPTS, 256, 256, convw[3], xc, 512L * NPTS, 256);

    // conv5: 512 -> 1024, then fused BN + max/mean pool
    gemm(convw[4], xc, 512L * NPTS, pre, 1024, NPTS, 512);
    bn_stats<<<1024, 256, 0, stream>>>(pre, 1024, meanb, rstdb);
    pool_maxmean<<<dim3(1024, BATCH), 256, 0, stream>>>(pre, meanb, rstdb, pbuf);

    // classifier head
    lin_bn<<<512, 256, 0, stream>>>(pbuf, lin1_w, lin1_b, h1, 2048, 512);
    lin_bn<<<256, 256, 0, stream>>>(h1, lin2_w, lin2_b, h2, 512, 256);
    lin_final<<<(320 + 63) / 64, 64, 0, stream>>>(h2, lin3_w, lin3_b, (float*)d_out);
}